// PoissonMCEstimator_5927054868974
// MI455X (gfx1250) — compile-verified
//
#include <hip/hip_runtime.h>

// CDNA5 / gfx1250 wave32 kernel for the regularized 3D Poisson Green's
// function MC estimator.
//
//  - r^2 tiles (16 queries x 16 landmarks) via V_WMMA_F32_16X16X4_F32:
//      A row = [-2x0, -2x1, -2x2, 1], B col = [y0, y1, y2, |y|^2],
//      C     = |x|^2 + eps^2  (constant),  D = r^2 + eps^2
//      clamp: max(r^2,0)+eps^2 == max(D, eps^2), done as a single v_max_i32
//      on the float bit pattern (exact for non-NaN D, which holds here).
//  - Landmark staging to LDS via the Tensor Data Mover (tensor_load_to_lds,
//    s_wait_tensorcnt), then a one-time repack that precomputes |y|^2.
//  - Tail is split into three unrolled phases (8x clamp, 8x rsq, 8x
//    accumulate) so each clamp->rsq and rsq->use dependency is >= 8
//    instructions apart: the TRANS pipe streams with no v_nop padding and
//    the FMA field dual-issues densely.

typedef __attribute__((ext_vector_type(2))) float        v2f;
typedef __attribute__((ext_vector_type(8))) float        v8f;
typedef __attribute__((ext_vector_type(4))) unsigned int v4u;
typedef __attribute__((ext_vector_type(4))) int          v4i;
typedef __attribute__((ext_vector_type(8))) int          v8i;

#define EPS2      (0.01f * 0.01f)
#define EPS2_BITS 0x38d1b717                 // bit pattern of 1e-4f
#define INV_4PI   0.07957747154594767f
#define MAX_M     2048

// ---- Tensor Data Mover: 1D copy of `ndwords` DWORDs from global to LDS ----
// D# fields per CDNA5 ISA sec 8.3/8.4: group0 = {count=1, lds_addr,
// global_addr, type=2}; group1 = {data_size=4B, tensor_dim0=ndwords,
// tensor_dim1=1, tile_dim0=ndwords, tile_dim1=1, dim0_stride=ndwords}.
__device__ __forceinline__ void tdm_load_1d(const void* gptr, unsigned lds_off,
                                            unsigned ndwords) {
    unsigned long long ga = (unsigned long long)(size_t)gptr;
    v4u g0;
    g0[0] = 1u;                                         // count=1, user D#
    g0[1] = lds_off;                                    // lds_addr
    g0[2] = (unsigned)(ga & 0xffffffffu);               // global_addr[31:0]
    g0[3] = (unsigned)((ga >> 32) & 0x01ffffffu)        // global_addr[56:32]
            | (2u << 30);                               // type=2 ("image")
    v8i g1;
    g1[0] = (int)(2u << 16);                            // wg_mask=0, data_size=4B
    g1[1] = (int)((ndwords & 0xffffu) << 16);           // tensor_dim0[15:0]
    g1[2] = (int)((ndwords >> 16) & 0xffffu) | (1 << 16); // dim0[31:16], dim1=1
    g1[3] = (int)((ndwords & 0xffffu) << 16);           // tile_dim0 = ndwords
    g1[4] = 1;                                          // tile_dim1=1, tile_dim2=0
    g1[5] = (int)ndwords;                               // dim0_stride[31:0]
    g1[6] = 0;                                          // stride hi / dim1_stride
    g1[7] = 0;
    v4i z4 = {};
#if __clang_major__ >= 23
    v8i z8 = {};
    __builtin_amdgcn_tensor_load_to_lds(g0, g1, z4, z4, z8, 0);
#else
    __builtin_amdgcn_tensor_load_to_lds(g0, g1, z4, z4, 0);
#endif
}

__global__ __launch_bounds__(256) void PoissonMC_wmma_kernel(
    const float* __restrict__ xq,    // (B,3)
    const float* __restrict__ xl,    // (M,3)
    const float* __restrict__ gl,    // (M,)
    float* __restrict__ out_v,       // (B,)
    float* __restrict__ out_g,       // (B,3)
    int M)
{
    __shared__ float  lds_g[MAX_M];          //  8 KB : g (TDM destination, final)
    __shared__ float  lds_stage[3 * MAX_M];  // 24 KB : raw x_land (TDM destination)
    __shared__ float4 lds_pack[MAX_M];       // 32 KB : {y0, y1, y2, |y|^2}

    // ---- async staging: wave 0 issues both DMAs, waits on TENSORcnt ----
    if (threadIdx.x < 32) {
        tdm_load_1d(xl, (unsigned)(size_t)(void*)lds_stage, 3u * (unsigned)M);
        tdm_load_1d(gl, (unsigned)(size_t)(void*)lds_g, (unsigned)M);
        __builtin_amdgcn_s_wait_tensorcnt(0);
    }
    __syncthreads();

    // ---- one-time repack: precompute |y|^2 so the hot loop skips it ----
    for (int i = threadIdx.x; i < M; i += 256) {
        float y0 = lds_stage[3 * i + 0];
        float y1 = lds_stage[3 * i + 1];
        float y2 = lds_stage[3 * i + 2];
        float4 P;
        P.x = y0; P.y = y1; P.z = y2;
        P.w = fmaf(y0, y0, fmaf(y1, y1, y2 * y2));
        lds_pack[i] = P;
    }
    __syncthreads();

    const int lane = threadIdx.x & 31;
    const int wave = threadIdx.x >> 5;       // 0..7
    const int col  = lane & 15;              // N column / own query row-in-tile
    const int hi   = lane >> 4;              // 0: lanes 0-15, 1: lanes 16-31
    const int tile = blockIdx.x * 8 + wave;  // 16-query tile index
    const int qrow = tile * 16 + col;        // this lane's own query index

    const float xq0 = xq[3 * qrow + 0];
    const float xq1 = xq[3 * qrow + 1];
    const float xq2 = xq[3 * qrow + 2];
    const float qn  = fmaf(xq0, xq0, fmaf(xq1, xq1, xq2 * xq2)) + EPS2;

    // A fragment: 16x4 f32. lanes 0-15: (K=0,K=1); lanes 16-31: (K=2,K=3)
    v2f afrag;
    afrag.x = hi ? (-2.0f * xq2) : (-2.0f * xq0);
    afrag.y = hi ? 1.0f          : (-2.0f * xq1);

    // C fragment: |x(row)|^2 + eps^2; VGPR i, lane l -> row i + 8*(l>=16)
    v8f cfrag;
#pragma unroll
    for (int i = 0; i < 8; ++i)
        cfrag[i] = __shfl(qn, i + (hi << 3), 32);

    v8f accV = {}, accC = {}, accX = {}, accY = {}, accZ = {};

    // ---- main loop, prefetch of next chunk's landmark data ----
    float4 P  = lds_pack[col];
    float  gc = lds_g[col];
    for (int mb = 0; mb < M; mb += 16) {
        const int nn = (mb + 16 < M) ? (mb + 16) : 0;   // next chunk (wraps)
        float4 Pn = lds_pack[nn + col];
        float  gn = lds_g[nn + col];

        // B fragment: 4x16 f32. lanes 0-15: rows K=0,1; lanes 16-31: K=2,3
        v2f bfrag;
        bfrag.x = hi ? P.z : P.x;
        bfrag.y = hi ? P.w : P.y;

        v8f d = __builtin_amdgcn_wmma_f32_16x16x4_f32(
            /*neg_a=*/false, afrag, /*neg_b=*/false, bfrag,
            /*c_mod=*/(short)0, cfrag, /*reuse_a=*/false, /*reuse_b=*/false);

        // phase 1: 8x clamp. max(d, EPS2) as signed-int max on the bit
        // pattern: exact for non-NaN d (negative floats have negative int
        // patterns; EPS2 > 0). Single v_max_i32 per element.
        v8f tt;
#pragma unroll
        for (int i = 0; i < 8; ++i) {
            int di = __float_as_int(d[i]);
            tt[i]  = __int_as_float((di > EPS2_BITS) ? di : EPS2_BITS);
        }

        // phase 2: 8x rsq, batched back-to-back on the TRANS pipe
        v8f rs;
#pragma unroll
        for (int i = 0; i < 8; ++i)
            rs[i] = __builtin_amdgcn_rsqf(tt[i]);        // v_rsq_f32

        // phase 3: accumulate (dense dual-issue FMA field)
        const float g = gc;
#pragma unroll
        for (int i = 0; i < 8; ++i) {
            float r1 = rs[i];
            accV[i]  = fmaf(g, r1, accV[i]);
            float s3 = r1 * r1 * r1;                     // t^(-3/2)
            float c  = g * s3;
            accC[i] += c;
            accX[i]  = fmaf(c, P.x, accX[i]);
            accY[i]  = fmaf(c, P.y, accY[i]);
            accZ[i]  = fmaf(c, P.z, accZ[i]);
        }
        P  = Pn;
        gc = gn;
    }

    // ---- reduce across the 16 lanes of each half (columns of the tile) ----
#pragma unroll
    for (int m = 1; m <= 8; m <<= 1) {
#pragma unroll
        for (int i = 0; i < 8; ++i) {
            accV[i] += __shfl_xor(accV[i], m, 32);
            accC[i] += __shfl_xor(accC[i], m, 32);
            accX[i] += __shfl_xor(accX[i], m, 32);
            accY[i] += __shfl_xor(accY[i], m, 32);
            accZ[i] += __shfl_xor(accZ[i], m, 32);
        }
    }

    // ---- store: lane i writes row i; lane 24+i writes row 8+i.
    // Both writer lanes' own (xq0,xq1,xq2) are exactly that row's coords.
    const float vs = INV_4PI / (float)M;
    const float gs = -INV_4PI / (float)M;
#pragma unroll
    for (int i = 0; i < 8; ++i) {
        if ((lane == i) || (lane == 24 + i)) {
            int r = tile * 16 + i + (hi << 3);
            out_v[r]         = vs * accV[i];
            out_g[3 * r + 0] = gs * (xq0 * accC[i] - accX[i]);
            out_g[3 * r + 1] = gs * (xq1 * accC[i] - accY[i]);
            out_g[3 * r + 2] = gs * (xq2 * accC[i] - accZ[i]);
        }
    }
}

extern "C" void kernel_launch(void* const* d_in, const int* in_sizes, int n_in,
                              void* d_out, int out_size, void* d_ws, size_t ws_size,
                              hipStream_t stream) {
    const float* xq = (const float*)d_in[0];   // (B,3)
    const float* xl = (const float*)d_in[1];   // (M,3)
    const float* gl = (const float*)d_in[2];   // (M,)
    const int B = in_sizes[0] / 3;             // 32768
    const int M = in_sizes[2];                 // 2048

    float* out_v = (float*)d_out;              // (B,)
    float* out_g = out_v + B;                  // (B,3)

    const int tiles  = B / 16;                 // 2048 wave-tiles
    const int blocks = tiles / 8;              // 8 waves (256 thr) per block

    PoissonMC_wmma_kernel<<<blocks, 256, 0, stream>>>(xq, xl, gl, out_v, out_g, M);
}